// Class_Capsule_15315853377614
// MI455X (gfx1250) — compile-verified
//
#include <hip/hip_runtime.h>
#include <cstdint>
#include <cstddef>

// Problem constants (from reference): x [32,16,16,16,8] -> u [32,4096,8]
// W [4096,10,8,16] ; out v [32,10,16] f32
#define NB   32      // batch
#define NN   4096    // input capsules
#define NCAP 10      // output capsules
#define DV   16      // capsule dim
#define DIN  8       // input dim
#define CD   (NCAP * DV)   // 160
#define NCHUNK 32
#define CHN  (NN / NCHUNK) // 128 n's per routing block

typedef float v2f __attribute__((ext_vector_type(2)));
typedef float v8f __attribute__((ext_vector_type(8)));

// ---------------------------------------------------------------------------
// Kernel 1: u_hat[n][b][c*16+d] = sum_i u[b][n][i] * W[n][c][i][d]
// One wave32 per n. Per-n GEMM [32x8]x[8x160] with V_WMMA_F32_16X16X4_F32.
// A (16x4 f32): lane holds row m=lane%16; VGPR j holds K = j + 2*(lane/16).
// B (4x16 f32): lane holds col n=lane%16; VGPR j holds K = j + 2*(lane/16).
// C/D (16x16 f32): VGPR v holds M = v + 8*(lane/16), N = lane%16.
// ---------------------------------------------------------------------------
__global__ __launch_bounds__(256) void capsule_uhat_wmma(
    const float* __restrict__ x, const float* __restrict__ W,
    float* __restrict__ uhat) {
  const int lane = threadIdx.x & 31;
  const int n    = blockIdx.x * 8 + (threadIdx.x >> 5);
  const int half = lane >> 4;   // 0: lanes 0-15, 1: lanes 16-31
  const int r    = lane & 15;   // M for A, N(=d) for B/C/D

  // A tiles: mt = batch tile (0..1), kc = K step (0..1)
  v2f A[2][2];
#pragma unroll
  for (int mt = 0; mt < 2; ++mt) {
#pragma unroll
    for (int kc = 0; kc < 2; ++kc) {
      const int b = mt * 16 + r;
      const int i = kc * 4 + 2 * half;
      const float* p = x + ((size_t)b * NN + n) * DIN + i;
      A[mt][kc].x = p[0];
      A[mt][kc].y = p[1];
    }
  }

  // B tiles: one 4x16 tile per capsule c per K step
  v2f Bm[NCAP][2];
#pragma unroll
  for (int c = 0; c < NCAP; ++c) {
#pragma unroll
    for (int kc = 0; kc < 2; ++kc) {
      const int i0 = kc * 4 + 2 * half;
      const float* p = W + (((size_t)n * NCAP + c) * DIN + i0) * DV + r;
      Bm[c][kc].x = p[0];    // K = i0
      Bm[c][kc].y = p[DV];   // K = i0 + 1
    }
  }

  float* outn = uhat + (size_t)n * (NB * CD);
#pragma unroll
  for (int mt = 0; mt < 2; ++mt) {
#pragma unroll
    for (int c = 0; c < NCAP; ++c) {
      v8f acc = {0.f, 0.f, 0.f, 0.f, 0.f, 0.f, 0.f, 0.f};
      acc = __builtin_amdgcn_wmma_f32_16x16x4_f32(
          false, A[mt][0], false, Bm[c][0], (short)0, acc, false, false);
      acc = __builtin_amdgcn_wmma_f32_16x16x4_f32(
          false, A[mt][1], false, Bm[c][1], (short)0, acc, false, false);
#pragma unroll
      for (int v = 0; v < 8; ++v) {
        const int b = mt * 16 + v + 8 * half;
        outn[(size_t)b * CD + c * DV + r] = acc[v];
      }
    }
  }
}

// ---------------------------------------------------------------------------
// Kernel R1: one routing iteration's logit update + softmax + partial N-sum.
// Block = (chunk of 128 n's, batch b). u_hat reads come from L2 (84MB < 192MB).
// iter==0: coefficients are uniform 0.1 (softmax of zeros over C=10).
// iter>=1: blog[b,n,c] = prev + dot_d(u_hat, v_prev); softmax over c.
// blog is fully rewritten at iter==1 before being read at iter==2 ->
// deterministic across replays without zero-init.
// ---------------------------------------------------------------------------
__global__ __launch_bounds__(256) void capsule_routing_accum(
    const float* __restrict__ uhat, float* __restrict__ blog,
    const float* __restrict__ vin, float* __restrict__ partial, int iter) {
  const int b     = blockIdx.y;
  const int chunk = blockIdx.x;
  const int n0    = chunk * CHN;
  const int t     = threadIdx.x;

  __shared__ float s_coeff[CHN * NCAP];  // logits, then softmax coeffs
  __shared__ float s_v[CD];

  if (iter > 0) {
    if (t < CD) s_v[t] = vin[b * CD + t];
    __syncthreads();
    // Stage 1: logit update, 1280 (n,c) tasks over 256 threads
    for (int task = t; task < CHN * NCAP; task += 256) {
      const int nl = task / NCAP;
      const int c  = task - nl * NCAP;
      const int n  = n0 + nl;
      const float* up = uhat + ((size_t)n * NB + b) * CD + c * DV;
      float dot = 0.f;
#pragma unroll
      for (int d = 0; d < DV; ++d) dot += up[d] * s_v[c * DV + d];
      const size_t bi = ((size_t)b * NN + n) * NCAP + c;
      const float prev = (iter == 1) ? 0.f : blog[bi];
      const float nb = prev + dot;
      blog[bi] = nb;
      s_coeff[task] = nb;
    }
    __syncthreads();
    // Softmax over c for each n (threads 0..127, one n each)
    if (t < CHN) {
      float lg[NCAP];
      float mx = -3.4e38f;
#pragma unroll
      for (int c = 0; c < NCAP; ++c) {
        lg[c] = s_coeff[t * NCAP + c];
        mx = fmaxf(mx, lg[c]);
      }
      float sum = 0.f;
#pragma unroll
      for (int c = 0; c < NCAP; ++c) {
        lg[c] = expf(lg[c] - mx);
        sum += lg[c];
      }
      const float inv = 1.f / sum;
#pragma unroll
      for (int c = 0; c < NCAP; ++c) s_coeff[t * NCAP + c] = lg[c] * inv;
    }
    __syncthreads();
  } else {
    for (int i = t; i < CHN * NCAP; i += 256) s_coeff[i] = 0.1f;
    __syncthreads();
  }

  // Stage 2: partial[b,chunk,c,d] = sum_{n in chunk} coeff[n,c]*u_hat[n,b,c,d]
  if (t < CD) {
    const int c = t >> 4;
    float acc = 0.f;
    for (int nl = 0; nl < CHN; ++nl) {
      acc += s_coeff[nl * NCAP + c] *
             uhat[((size_t)(n0 + nl) * NB + b) * CD + t];
    }
    partial[((size_t)b * NCHUNK + chunk) * CD + t] = acc;
  }
}

// ---------------------------------------------------------------------------
// Kernel R2: reduce chunk partials -> s[b,c,d], squash over d, emit v.
// squash(s) = (s2/(1+s2)/sqrt(s2+1e-7)) * s, s2 = sum_d s^2 per (b,c).
// ---------------------------------------------------------------------------
__global__ __launch_bounds__(160) void capsule_reduce_squash(
    const float* __restrict__ partial, float* __restrict__ vout,
    float* __restrict__ out, int last) {
  const int b = blockIdx.x;
  const int t = threadIdx.x;   // 0..159 = (c,d)
  const int c = t >> 4;

  float s = 0.f;
#pragma unroll
  for (int ch = 0; ch < NCHUNK; ++ch)
    s += partial[((size_t)b * NCHUNK + ch) * CD + t];

  __shared__ float sh[CD];
  __shared__ float s2sh[NCAP];
  sh[t] = s * s;
  __syncthreads();
  if ((t & 15) == 0) {
    float s2 = 0.f;
#pragma unroll
    for (int k = 0; k < DV; ++k) s2 += sh[c * DV + k];
    s2sh[c] = s2;
  }
  __syncthreads();
  const float s2 = s2sh[c];
  const float scale = s2 / ((1.f + s2) * sqrtf(s2 + 1e-7f));
  const float vv = scale * s;
  vout[b * CD + t] = vv;
  if (last) out[b * CD + t] = vv;
}

// ---------------------------------------------------------------------------
extern "C" void kernel_launch(void* const* d_in, const int* in_sizes, int n_in,
                              void* d_out, int out_size, void* d_ws,
                              size_t ws_size, hipStream_t stream) {
  (void)in_sizes; (void)n_in; (void)out_size; (void)ws_size;
  const float* x = (const float*)d_in[0];   // [32,16,16,16,8] == [32,4096,8]
  const float* W = (const float*)d_in[1];   // [4096,10,8,16]
  float* out = (float*)d_out;               // [32,10,16]

  float* ws      = (float*)d_ws;
  float* uhat    = ws;                                      // 20,971,520 f32
  float* blog    = uhat + (size_t)NN * NB * CD;             //  1,310,720 f32
  float* partial = blog + (size_t)NB * NN * NCAP;           //    163,840 f32
  float* vbuf    = partial + (size_t)NB * NCHUNK * CD;      //      5,120 f32

  capsule_uhat_wmma<<<NN / 8, 256, 0, stream>>>(x, W, uhat);
  for (int it = 0; it < 3; ++it) {
    capsule_routing_accum<<<dim3(NCHUNK, NB), 256, 0, stream>>>(
        uhat, blog, vbuf, partial, it);
    capsule_reduce_squash<<<NB, 160, 0, stream>>>(partial, vbuf, out,
                                                  it == 2 ? 1 : 0);
  }
}